// EuclideanDistance_1580547968539
// MI455X (gfx1250) — compile-verified
//
#include <hip/hip_runtime.h>

typedef __attribute__((ext_vector_type(2))) float v2f;
typedef __attribute__((ext_vector_type(8))) float v8f;

#define KDIM 256

// ---------------------------------------------------------------------------
// Prologue: per-row squared norms for mat_1 (n1 rows) and mat_2 (n2 rows).
// One thread per row, float4 loads (row = 1 KB = 64 float4s). 24 MB total read.
// ---------------------------------------------------------------------------
__global__ __launch_bounds__(256) void row_norms_kernel(
    const float* __restrict__ m1, const float* __restrict__ m2,
    float* __restrict__ norms, int n1, int n2) {
  int r = blockIdx.x * blockDim.x + threadIdx.x;
  if (r >= n1 + n2) return;
  const float* src = (r < n1) ? (m1 + (size_t)r * KDIM)
                              : (m2 + (size_t)(r - n1) * KDIM);
  const float4* p = (const float4*)src;
  float s = 0.0f;
#pragma unroll 8
  for (int i = 0; i < KDIM / 4; ++i) {
    float4 v = p[i];
    s += v.x * v.x + v.y * v.y + v.z * v.z + v.w * v.w;
  }
  norms[r] = s;
}

// ---------------------------------------------------------------------------
// Main kernel: D = sq1[r] + sq2[c] - 2 * (A · B^T)
// Block = 256 threads = 8 waves arranged 2x4; block tile = 128x128.
// Each wave: 64x32 output = 4x2 grid of 16x16 WMMA tiles, K-loop step 4 using
// v_wmma_f32_16x16x4_f32.
//
// Fragment layouts (ISA 7.12.2, wave32):
//  A (16x4 f32, 2 VGPRs/lane): lane L(<16) row=L, holds K = {2*half, 2*half+1}
//  B (4x16 f32, 2 VGPRs/lane): lane L(<16) col=N=L, holds K = {2*half, 2*half+1}
//  -> B[k][n] = mat2[c0+n][k], so B fragment load == A fragment addressing.
//  C/D (16x16 f32, 8 VGPRs): vgpr i -> M = i + 8*half, N = lane&15.
// ---------------------------------------------------------------------------
__global__ __launch_bounds__(256) void dist_wmma_kernel(
    const float* __restrict__ A, const float* __restrict__ B,
    const float* __restrict__ sq1, const float* __restrict__ sq2,
    float* __restrict__ out, int ncols) {
  const int lane = threadIdx.x & 31;
  const int wid  = threadIdx.x >> 5;
  const int half = lane >> 4;     // 0: lanes 0-15, 1: lanes 16-31
  const int lrow = lane & 15;

  const int row0 = blockIdx.x * 128 + (wid >> 2) * 64;  // 4 M-tiles of 16
  const int col0 = blockIdx.y * 128 + (wid & 3) * 32;   // 2 N-tiles of 16

  v8f acc[4][2];
#pragma unroll
  for (int m = 0; m < 4; ++m)
#pragma unroll
    for (int n = 0; n < 2; ++n)
      acc[m][n] = (v8f){0.f, 0.f, 0.f, 0.f, 0.f, 0.f, 0.f, 0.f};

  // Per-lane fragment base pointers (8-byte aligned: k and 2*half are even).
  const float* aBase = A + (size_t)(row0 + lrow) * KDIM + 2 * half;
  const float* bBase = B + (size_t)(col0 + lrow) * KDIM + 2 * half;

  for (int k = 0; k < KDIM; k += 4) {
    v2f af[4], bf[2];
#pragma unroll
    for (int m = 0; m < 4; ++m)
      af[m] = *(const v2f*)(aBase + (size_t)m * 16 * KDIM + k);
#pragma unroll
    for (int n = 0; n < 2; ++n)
      bf[n] = *(const v2f*)(bBase + (size_t)n * 16 * KDIM + k);

#pragma unroll
    for (int m = 0; m < 4; ++m)
#pragma unroll
      for (int n = 0; n < 2; ++n)
        acc[m][n] = __builtin_amdgcn_wmma_f32_16x16x4_f32(
            /*neg_a=*/false, af[m], /*neg_b=*/false, bf[n],
            /*c_mod=*/(short)0, acc[m][n],
            /*reuse_a=*/false, /*reuse_b=*/false);
  }

  // Epilogue: out[r, c] = sq1[r] + sq2[c] - 2*acc
#pragma unroll
  for (int m = 0; m < 4; ++m) {
    const int rbase = row0 + m * 16 + half * 8;
#pragma unroll
    for (int n = 0; n < 2; ++n) {
      const int c = col0 + n * 16 + lrow;
      const float s2 = sq2[c];
      float* outp = out + (size_t)rbase * ncols + c;
#pragma unroll
      for (int i = 0; i < 8; ++i) {
        outp[(size_t)i * ncols] = sq1[rbase + i] + s2 - 2.0f * acc[m][n][i];
      }
    }
  }
}

// ---------------------------------------------------------------------------
extern "C" void kernel_launch(void* const* d_in, const int* in_sizes, int n_in,
                              void* d_out, int out_size, void* d_ws, size_t ws_size,
                              hipStream_t stream) {
  const float* m1 = (const float*)d_in[0];
  const float* m2 = (const float*)d_in[1];
  const int n1 = in_sizes[0] / KDIM;  // 16384
  const int n2 = in_sizes[1] / KDIM;  // 8192

  float* norms = (float*)d_ws;        // n1 + n2 floats (96 KB)
  float* sq1 = norms;
  float* sq2 = norms + n1;

  const int total = n1 + n2;
  row_norms_kernel<<<(total + 255) / 256, 256, 0, stream>>>(m1, m2, norms, n1, n2);

  dim3 grid(n1 / 128, n2 / 128);      // 128 x 64 blocks
  dist_wmma_kernel<<<grid, dim3(256, 1, 1), 0, stream>>>(m1, m2, sq1, sq2,
                                                         (float*)d_out, n2);
}